// MyModel_61933428414456
// MI455X (gfx1250) — compile-verified
//
#include <hip/hip_runtime.h>
#include <stdint.h>

#define NUM_BINS      1000
#define BLOCK_THREADS 256
#define WAVES_PER_BLK 8
#define CHUNK_FLOATS  512          // floats per wave per buffer
#define CHUNK_BYTES   2048

// ---------------------------------------------------------------------------
// Zero the global histogram (workspace is poisoned; must re-zero every call).
// ---------------------------------------------------------------------------
__global__ void hist_zero_kernel(unsigned int* __restrict__ g_hist) {
    unsigned t = threadIdx.x;
    if (t < 1024u) g_hist[t] = 0u;
}

// ---------------------------------------------------------------------------
// Main histogram kernel: async global->LDS double-buffered streaming,
// LDS-privatized histogram, block-level flush via global atomics.
// ---------------------------------------------------------------------------
__global__ __launch_bounds__(BLOCK_THREADS)
void hist_main_kernel(const float* __restrict__ x, unsigned long long n,
                      unsigned int* __restrict__ g_hist, unsigned int nchunks) {
    __shared__ unsigned int lhist[1024];
    __shared__ float stage[WAVES_PER_BLK][2][CHUNK_FLOATS];

    for (int i = threadIdx.x; i < 1024; i += BLOCK_THREADS) lhist[i] = 0u;
    __syncthreads();

    const unsigned lane    = threadIdx.x & 31u;
    const unsigned wave    = threadIdx.x >> 5;
    const unsigned gwave   = blockIdx.x * WAVES_PER_BLK + wave;
    const unsigned wstride = gridDim.x * WAVES_PER_BLK;

    // LDS byte offsets of this wave's two staging buffers (generic->LDS trunc)
    const uint32_t lds0 = (uint32_t)(uintptr_t)(&stage[wave][0][0]) + lane * 16u;
    const uint32_t lds1 = (uint32_t)(uintptr_t)(&stage[wave][1][0]) + lane * 16u;

    // Issue 4 x b128 async copies for one 2KB chunk. INST_OFFSET is applied to
    // both the LDS destination and the global source address (ISA §15.18), so
    // one (vdst, vaddr) pair covers all four stripes.
    auto issue = [&](uint32_t lds_base, unsigned chunk) {
        uint32_t goff = chunk * (uint32_t)CHUNK_BYTES + lane * 16u;
        asm volatile(
            "global_load_async_to_lds_b128 %0, %1, %2\n\t"
            "global_load_async_to_lds_b128 %0, %1, %2 offset:512\n\t"
            "global_load_async_to_lds_b128 %0, %1, %2 offset:1024\n\t"
            "global_load_async_to_lds_b128 %0, %1, %2 offset:1536"
            :
            : "v"(lds_base), "v"(goff), "s"(x)
            : "memory");
    };

    // Consume one staged 2KB chunk: 16 floats per lane -> 16 LDS atomics.
    auto process = [&](const float* buf) {
#pragma unroll
        for (int j = 0; j < 4; ++j) {
            float4 v = *(const float4*)(buf + j * 128 + lane * 4);
            atomicAdd(&lhist[((unsigned)(int)v.x) & 1023u], 1u);
            atomicAdd(&lhist[((unsigned)(int)v.y) & 1023u], 1u);
            atomicAdd(&lhist[((unsigned)(int)v.z) & 1023u], 1u);
            atomicAdd(&lhist[((unsigned)(int)v.w) & 1023u], 1u);
        }
    };

    unsigned chunk = gwave;
    int cur = 0;
    if (chunk < nchunks) issue(lds0, chunk);
    while (chunk < nchunks) {
        unsigned next = chunk + wstride;
        if (next < nchunks) {
            issue(cur ? lds0 : lds1, next);
            // 8 outstanding; first 4 (current buffer) complete in order
            asm volatile("s_wait_asynccnt 0x4" ::: "memory");
        } else {
            asm volatile("s_wait_asynccnt 0x0" ::: "memory");
        }
        process(cur ? &stage[wave][1][0] : &stage[wave][0][0]);
        cur ^= 1;
        chunk = next;
    }

    // Tail elements (n not multiple of 512): handled directly by block 0.
    if (blockIdx.x == 0) {
        for (unsigned long long i =
                 (unsigned long long)nchunks * (unsigned long long)CHUNK_FLOATS + threadIdx.x;
             i < n; i += BLOCK_THREADS) {
            atomicAdd(&lhist[((unsigned)(int)x[i]) & 1023u], 1u);
        }
    }

    __syncthreads();
    for (int i = threadIdx.x; i < NUM_BINS; i += BLOCK_THREADS) {
        unsigned c = lhist[i];
        if (c) atomicAdd(&g_hist[i], c);
    }
}

// ---------------------------------------------------------------------------
// Compaction: sorted unique values + counts, zero-padded to NUM_BINS each.
// out[0..999] = unique values (float), out[1000..1999] = counts (float).
// ---------------------------------------------------------------------------
__global__ __launch_bounds__(1024)
void hist_finalize_kernel(const unsigned int* __restrict__ g_hist,
                          float* __restrict__ out) {
    __shared__ unsigned int flags[1024];
    const unsigned t = threadIdx.x;

    const unsigned c = (t < NUM_BINS) ? g_hist[t] : 0u;
    flags[t] = (c != 0u) ? 1u : 0u;
    if (t < NUM_BINS) {          // fill_value = 0 for values, 0 for counts
        out[t] = 0.0f;
        out[NUM_BINS + t] = 0.0f;
    }
    __threadfence_block();
    __syncthreads();

    // Hillis-Steele inclusive scan over 1024 flags
    for (int off = 1; off < 1024; off <<= 1) {
        unsigned add = (t >= (unsigned)off) ? flags[t - off] : 0u;
        __syncthreads();
        flags[t] += add;
        __syncthreads();
    }

    if (t < NUM_BINS && c != 0u) {
        unsigned idx = flags[t] - 1u;   // exclusive position
        out[idx] = (float)t;
        out[NUM_BINS + idx] = (float)c;
    }
}

// ---------------------------------------------------------------------------
extern "C" void kernel_launch(void* const* d_in, const int* in_sizes, int n_in,
                              void* d_out, int out_size, void* d_ws, size_t ws_size,
                              hipStream_t stream) {
    const float* x = (const float*)d_in[0];
    const unsigned long long n = (unsigned long long)in_sizes[0];
    unsigned int* g_hist = (unsigned int*)d_ws;
    float* out = (float*)d_out;

    const unsigned nchunks = (unsigned)(n / (unsigned long long)CHUNK_FLOATS);

    hist_zero_kernel<<<1, 1024, 0, stream>>>(g_hist);

    unsigned blocks = 1024;
    if (nchunks < blocks * WAVES_PER_BLK) {
        blocks = (nchunks + WAVES_PER_BLK - 1) / WAVES_PER_BLK;
        if (blocks == 0) blocks = 1;
    }
    hist_main_kernel<<<blocks, BLOCK_THREADS, 0, stream>>>(x, n, g_hist, nchunks);

    hist_finalize_kernel<<<1, 1024, 0, stream>>>(g_hist, out);
}